// ModelWrapper_29085518528809
// MI455X (gfx1250) — compile-verified
//
#include <hip/hip_runtime.h>

// YOLO box decode, MI455X (gfx1250).
// Memory-bound: ~147 MB moved -> ~6.3 us floor at 23.3 TB/s.
// Strategy: block-cooperative staging with CDNA5 async global<->LDS b128 copies
// (ASYNCcnt pipeline, no VGPR round trip), NT temporal hints (pure streaming,
// no reuse -> don't churn the 192MB L2), conflict-free LDS row reads
// (odd 85-float stride), LDS-staged coalesced output drain.

#define C_IN            85
#define C_OUT           6
#define ROWS_PER_BLOCK  256
#define THREADS         256
#define CONF_THRESH_F   0.25f

// Low 32 bits of a generic pointer into __shared__ memory == LDS byte offset
// within the workgroup allocation (generic LDS addr = {aperture_hi, offset_lo}).
__device__ __forceinline__ unsigned lds_off(const void* p) {
    return (unsigned)(unsigned long long)p;
}

__device__ __forceinline__ void async_load_b128_to_lds(unsigned loff, const float* gaddr) {
    asm volatile("global_load_async_to_lds_b128 %0, %1, off th:TH_LOAD_NT"
                 :: "v"(loff), "v"(gaddr)
                 : "memory");
}

__device__ __forceinline__ void async_store_b128_from_lds(float* gaddr, unsigned loff) {
    asm volatile("global_store_async_from_lds_b128 %0, %1, off th:TH_STORE_NT"
                 :: "v"(gaddr), "v"(loff)
                 : "memory");
}

__device__ __forceinline__ void wait_asynccnt0() {
    asm volatile("s_wait_asynccnt 0" ::: "memory");
}

__global__ void __launch_bounds__(THREADS)
yolo_decode_kernel(const float* __restrict__ in, float* __restrict__ out, int rows) {
    __shared__ float s_in [ROWS_PER_BLOCK * C_IN];   // 87,040 B
    __shared__ float s_out[ROWS_PER_BLOCK * C_OUT];  //  6,144 B

    const int tid     = (int)threadIdx.x;
    const int rowBase = (int)blockIdx.x * ROWS_PER_BLOCK;
    int validRows = rows - rowBase;
    if (validRows <= 0) return;
    if (validRows > ROWS_PER_BLOCK) validRows = ROWS_PER_BLOCK;

    // ---------- Phase 1: async copy global -> LDS (coalesced b128 chunks) ----------
    // Block region starts at rowBase*85*4 bytes = blockIdx*87040 B -> always 16B aligned.
    const float* gsrc     = in + (size_t)rowBase * C_IN;
    const int    inFloats = validRows * C_IN;
    const int    fullIn   = inFloats >> 2;           // # of 16B chunks
    for (int i = tid; i < fullIn; i += THREADS) {
        async_load_b128_to_lds(lds_off(&s_in[i * 4]), gsrc + i * 4);
    }
    // Tail floats (only possible in a partial last block): plain copy.
    for (int f = (fullIn << 2) + tid; f < inFloats; f += THREADS) {
        s_in[f] = gsrc[f];
    }
    wait_asynccnt0();        // this wave's async LDS writes have landed
    __syncthreads();         // all waves' writes visible

    // ---------- Phase 2: one row per thread, conflict-free LDS reads ----------
    if (tid < validRows) {
        const float* r = &s_in[tid * C_IN];          // stride 85 floats: odd -> no bank conflicts
        const float x = r[0], y = r[1], w = r[2], h = r[3], conf = r[4];

        float best = r[5];
        int   bestIdx = 0;
        #pragma unroll
        for (int k = 1; k < C_IN - 5; ++k) {
            const float v = r[5 + k];
            if (v > best) { best = v; bestIdx = k; }  // strict '>' == first-occurrence argmax
        }

        const float score = conf * best;
        const bool  keep  = score > CONF_THRESH_F;
        const float hw = 0.5f * w, hh = 0.5f * h;

        float* ro = &s_out[tid * C_OUT];
        ro[0] = keep ? (x - hw)        : 0.0f;
        ro[1] = keep ? (y - hh)        : 0.0f;
        ro[2] = keep ? (x + hw)        : 0.0f;
        ro[3] = keep ? (y + hh)        : 0.0f;
        ro[4] = keep ? score           : 0.0f;
        ro[5] = keep ? (float)bestIdx  : 0.0f;
    }
    __syncthreads();

    // ---------- Phase 3: async store LDS -> global (coalesced b128 chunks) ----------
    float*    gdst      = out + (size_t)rowBase * C_OUT;  // blockIdx*6144 B -> 16B aligned
    const int outFloats = validRows * C_OUT;
    const int fullOut   = outFloats >> 2;
    for (int i = tid; i < fullOut; i += THREADS) {
        async_store_b128_from_lds(gdst + i * 4, lds_off(&s_out[i * 4]));
    }
    for (int f = (fullOut << 2) + tid; f < outFloats; f += THREADS) {
        gdst[f] = s_out[f];
    }
    wait_asynccnt0();   // drain before wave retirement (S_ENDPGM wait-idle also covers this)
}

extern "C" void kernel_launch(void* const* d_in, const int* in_sizes, int n_in,
                              void* d_out, int out_size, void* d_ws, size_t ws_size,
                              hipStream_t stream) {
    (void)n_in; (void)out_size; (void)d_ws; (void)ws_size;
    const float* in  = (const float*)d_in[0];
    float*       out = (float*)d_out;
    const int rows   = in_sizes[0] / C_IN;                   // 16*25200 = 403,200
    const int blocks = (rows + ROWS_PER_BLOCK - 1) / ROWS_PER_BLOCK;  // 1575
    yolo_decode_kernel<<<blocks, THREADS, 0, stream>>>(in, out, rows);
}